// Memory_21818433864466
// MI455X (gfx1250) — compile-verified
//
#include <hip/hip_runtime.h>
#include <hip/hip_bf16.h>
#include <stdint.h>

// Problem geometry (fixed by the reference)
#define WNN_B        256
#define WNN_INBITS   1024
#define WNN_WORDS    32        // 1024 bits / 32 per u32
#define WNN_NEURONS  4096
#define WNN_MEMSZ    65536     // 2^16

typedef __attribute__((address_space(3))) uint32_t lds_u32;

// ---------------------------------------------------------------------------
// Kernel 1: pack the [256 x 1024] int32 bool tensor into [256 x 32] u32 masks.
// wave32-native: one ballot per wave packs 32 bools into one word.
// ---------------------------------------------------------------------------
__global__ void __launch_bounds__(256)
pack_bits_kernel(const int* __restrict__ bits, uint32_t* __restrict__ packed) {
  const uint32_t tid = blockIdx.x * 256u + threadIdx.x;   // one bit per thread
  const bool pred = bits[tid] != 0;
#if __has_builtin(__builtin_amdgcn_ballot_w32)
  const uint32_t mask = __builtin_amdgcn_ballot_w32(pred);
  if ((threadIdx.x & 31u) == 0u) packed[tid >> 5] = mask;
#else
  if ((threadIdx.x & 31u) == 0u) {
    uint32_t w = 0u;
    #pragma unroll
    for (int i = 0; i < 32; ++i) w |= (uint32_t)(bits[tid + i] != 0) << i;
    packed[tid >> 5] = w;
  }
#endif
}

// ---------------------------------------------------------------------------
// Kernel 2: per (batch, neuron) address build + 1 GiB table gather.
// Block = one batch row, 256 consecutive neurons. The batch's 128 B bitmask
// is staged into LDS with the gfx1250 async global->LDS path.
// ---------------------------------------------------------------------------
__global__ void __launch_bounds__(256)
wnn_lookup_kernel(const uint32_t* __restrict__ packed,
                  const int* __restrict__ connections,
                  const int* __restrict__ memory,
                  float* __restrict__ out) {
  __shared__ uint32_t sbits[WNN_WORDS];

  const uint32_t b = blockIdx.x >> 4;                          // batch   [0,256)
  const uint32_t n = ((blockIdx.x & 15u) << 8) | threadIdx.x;  // neuron  [0,4096)

  // --- stage this batch's 32 packed words into LDS (async, first wave) ---
  if (threadIdx.x < WNN_WORDS) {
    const uint32_t* gp = packed + b * WNN_WORDS + threadIdx.x;
#if defined(__gfx1250__)
    const uint32_t lds_off = (uint32_t)(uintptr_t)(lds_u32*)(&sbits[threadIdx.x]);
    asm volatile("global_load_async_to_lds_b32 %0, %1, off"
                 :
                 : "v"(lds_off), "v"(gp)
                 : "memory");
#else
    sbits[threadIdx.x] = *gp;
#endif
  }
#if defined(__gfx1250__)
#if __has_builtin(__builtin_amdgcn_s_wait_asynccnt)
  __builtin_amdgcn_s_wait_asynccnt(0);
#else
  asm volatile("s_wait_asynccnt 0x0" ::: "memory");
#endif
#endif
  __syncthreads();

  // --- load this neuron's 16 wire indices (coalesced 4 x b128 per thread) ---
  const int4* cp = reinterpret_cast<const int4*>(connections) + (size_t)n * 4;
  const int4 c0 = cp[0];
  const int4 c1 = cp[1];
  const int4 c2 = cp[2];
  const int4 c3 = cp[3];

  // --- build the 16-bit RAM address from the LDS bitmask ---
#define WNN_BIT(c) ((sbits[((uint32_t)(c)) >> 5] >> (((uint32_t)(c)) & 31u)) & 1u)
  uint32_t addr = 0u;
  addr |= WNN_BIT(c0.x) << 0;   addr |= WNN_BIT(c0.y) << 1;
  addr |= WNN_BIT(c0.z) << 2;   addr |= WNN_BIT(c0.w) << 3;
  addr |= WNN_BIT(c1.x) << 4;   addr |= WNN_BIT(c1.y) << 5;
  addr |= WNN_BIT(c1.z) << 6;   addr |= WNN_BIT(c1.w) << 7;
  addr |= WNN_BIT(c2.x) << 8;   addr |= WNN_BIT(c2.y) << 9;
  addr |= WNN_BIT(c2.z) << 10;  addr |= WNN_BIT(c2.w) << 11;
  addr |= WNN_BIT(c3.x) << 12;  addr |= WNN_BIT(c3.y) << 13;
  addr |= WNN_BIT(c3.z) << 14;  addr |= WNN_BIT(c3.w) << 15;
#undef WNN_BIT

  // --- single random gather from the 1 GiB table; NT: don't pollute L2 ---
  const size_t midx = ((size_t)n << 16) | (size_t)addr;
  const int v = __builtin_nontemporal_load(memory + midx);

  // --- coalesced output (harness stores integer outputs as float) ---
  out[((size_t)b << 12) | n] = (float)v;
}

// ---------------------------------------------------------------------------
// Launch wrapper
//   d_in[0]: input_bits  [256*1024]  int32 (0/1)
//   d_in[1]: connections [4096*16]   int32
//   d_in[2]: memory      [4096*65536] int32
//   d_out  : [256*4096] float
//   d_ws   : >= 32 KB scratch for packed bitmasks
// ---------------------------------------------------------------------------
extern "C" void kernel_launch(void* const* d_in, const int* in_sizes, int n_in,
                              void* d_out, int out_size, void* d_ws, size_t ws_size,
                              hipStream_t stream) {
  (void)in_sizes; (void)n_in; (void)out_size; (void)ws_size;

  const int* bits = (const int*)d_in[0];
  const int* conn = (const int*)d_in[1];
  const int* mem  = (const int*)d_in[2];
  float* out      = (float*)d_out;
  uint32_t* packed = (uint32_t*)d_ws;   // 256*32 u32 = 32 KB

  // 262144 bits -> 1024 blocks of 256 threads
  pack_bits_kernel<<<(WNN_B * WNN_INBITS) / 256, 256, 0, stream>>>(bits, packed);

  // 256 batches * 16 neuron-tiles = 4096 blocks of 256 threads
  wnn_lookup_kernel<<<WNN_B * (WNN_NEURONS / 256), 256, 0, stream>>>(packed, conn, mem, out);
}